// ContrastiveLabeledLoss_19396072308794
// MI455X (gfx1250) — compile-verified
//
#include <hip/hip_runtime.h>
#include <hip/hip_bf16.h>
#include <stdint.h>

#define NUM_CLASSES 64
#define D_DIM 256
#define A_PITCH 257                                   // dwords per anchor row in LDS (bank-conflict-free)
#define BINS_OFF (NUM_CLASSES * A_PITCH * 4)          // 65792 bytes
#define SMEM_BYTES (BINS_OFF + NUM_CLASSES * 4)       // + 64 float class bins

typedef __attribute__((ext_vector_type(2))) float v2f;
typedef __attribute__((ext_vector_type(8))) float v8f;

// ---------------- workspace layout (byte offsets into d_ws) ----------------
//    0 : int   anchor_idx[64]
//  256 : int   counts[64]
//  512 : float sums[64]
// 1024 : float anchors[64*256]   (64 KB)

// -------- CDNA5 async global->LDS (inline asm; portable across toolchains) --------
__device__ __forceinline__ void async_b32_to_lds(uint32_t lds_byte_off, const void* gaddr) {
    asm volatile("global_load_async_to_lds_b32 %0, %1, off"
                 :: "v"(lds_byte_off), "v"(gaddr)
                 : "memory");
}
__device__ __forceinline__ void wait_async0() {
    asm volatile("s_wait_asynccnt 0" ::: "memory");
}

// ---------------------------------------------------------------------------
__global__ void k_init(int* __restrict__ anchor_idx, int* __restrict__ counts,
                       float* __restrict__ sums) {
    int t = threadIdx.x;
    if (t < NUM_CLASSES) {
        anchor_idx[t] = 0x7fffffff;
        counts[t]     = 0;
        sums[t]       = 0.0f;
    }
}

__global__ void k_stats(const int* __restrict__ labels, int N,
                        int* __restrict__ anchor_idx, int* __restrict__ counts) {
    __shared__ int smin[NUM_CLASSES];
    __shared__ int scnt[NUM_CLASSES];
    int t = threadIdx.x;
    if (t < NUM_CLASSES) { smin[t] = 0x7fffffff; scnt[t] = 0; }
    __syncthreads();
    int i = blockIdx.x * blockDim.x + t;
    if (i < N) {
        int c = labels[i] & (NUM_CLASSES - 1);
        atomicMin(&smin[c], i);
        atomicAdd(&scnt[c], 1);
    }
    __syncthreads();
    if (t < NUM_CLASSES && scnt[t] > 0) {
        atomicMin(&anchor_idx[t], smin[t]);
        atomicAdd(&counts[t], scnt[t]);
    }
}

__global__ void k_gather(const float* __restrict__ X, const int* __restrict__ anchor_idx,
                         int N, float* __restrict__ anchors) {
    int c = blockIdx.x;
    int a = anchor_idx[c];
    if (a > N - 1) a = N - 1;                         // safe_anchor = min(idx, N-1)
    anchors[c * D_DIM + threadIdx.x] = X[(size_t)a * D_DIM + threadIdx.x];
}

// ---------------------------------------------------------------------------
// Main streaming pass. One wave32 handles a 16-row tile.
// A-matrix (16x4 fp32, per WMMA step k): lane m (0..15) supplies K=0,1 =
// squared diffs of cols {4k,4k+1} of row m; lane m+16 supplies K=2,3 =
// cols {4k+2,4k+3}. B = all-ones (layout-independent), so after the K loop
// acc[r] (lanes 0-15) = d2 of row r, acc[r] (lanes 16-31) = d2 of row 8+r,
// replicated across all columns -> zero-shuffle extraction.
__global__ void __launch_bounds__(256)
k_main(const float* __restrict__ X, const int* __restrict__ labels, int N,
       const float* __restrict__ anchors_g, float* __restrict__ sums) {
    extern __shared__ char smem[];
    float* bins = (float*)(smem + BINS_OFF);
    const int tid = threadIdx.x;

    // Stage the 64x256 anchor table into LDS with pitch 257 dwords (async path).
    #pragma unroll 4
    for (int i = 0; i < NUM_CLASSES; ++i) {
        async_b32_to_lds((uint32_t)(i * (A_PITCH * 4) + tid * 4),
                         anchors_g + i * D_DIM + tid);
    }
    if (tid < NUM_CLASSES) bins[tid] = 0.0f;
    wait_async0();
    __syncthreads();

    const int lane  = tid & 31;
    const int warp  = tid >> 5;
    const int m     = lane & 15;
    const int kpart = lane >> 4;                      // 0: K={0,1}, 1: K={2,3}
    const int rsel  = lane & 7;
    const bool writer = (lane < 8) || (lane >= 24);   // rows 0-7 / rows 8-15
    const int numTiles = N >> 4;

    v2f ones; ones[0] = 1.0f; ones[1] = 1.0f;

    for (int tile = blockIdx.x * 8 + warp; tile < numTiles; tile += gridDim.x * 8) {
        const int row = tile * 16 + m;
        const int l   = labels[row] & (NUM_CLASSES - 1);
        const float* xrow = X + (size_t)row * D_DIM + 2 * kpart;     // 8B aligned
        const float* arow = (const float*)(smem + l * (A_PITCH * 4)) + 2 * kpart;

        v8f acc = {};
        #pragma unroll 8
        for (int k = 0; k < 64; ++k) {
            const int cb = 4 * k;
            float2 xv = *(const float2*)(xrow + cb);                 // global b64
            float  a0 = arow[cb];                                    // ds b32, conflict-free
            float  a1 = arow[cb + 1];
            float  d0 = xv.x - a0;
            float  d1 = xv.y - a1;
            v2f av; av[0] = d0 * d0; av[1] = d1 * d1;
            acc = __builtin_amdgcn_wmma_f32_16x16x4_f32(
                      false, av, false, ones, (short)0, acc, false, false);
        }

        // Branchless 8-way component select in ALL lanes (v_cndmask chain,
        // no exec waterfall); only the atomic sits under divergence.
        float val = acc[0];
        #pragma unroll
        for (int r = 1; r < 8; ++r) val = (rsel == r) ? acc[r] : val;

        // Writers: lanes 0-7 -> rows 0-7 (acc[lane]); lanes 24-31 -> rows 8-15
        // (acc[lane-24]); both already hold the matching row's label l.
        // Anchor rows contribute exactly 0, matching the reference's exclusion.
        if (writer) {
            atomicAdd(&bins[l], val);                                // LDS f32 atomic
        }
    }
    __syncthreads();
    if (tid < NUM_CLASSES) atomicAdd(&sums[tid], bins[tid]);         // one global atomic/class/block
}

__global__ void k_final(const float* __restrict__ sums, const int* __restrict__ counts,
                        float* __restrict__ out) {
    __shared__ float red[NUM_CLASSES];
    int t = threadIdx.x;
    float cnt = (float)counts[t];
    float v = (cnt >= 2.0f) ? sums[t] / fmaxf(cnt - 1.0f, 1.0f) : 0.0f;
    red[t] = v;
    __syncthreads();
    if (t == 0) {
        float s = 0.0f;
        #pragma unroll
        for (int i = 0; i < NUM_CLASSES; ++i) s += red[i];
        out[0] = s;
    }
}

// ---------------------------------------------------------------------------
extern "C" void kernel_launch(void* const* d_in, const int* in_sizes, int n_in,
                              void* d_out, int out_size, void* d_ws, size_t ws_size,
                              hipStream_t stream) {
    const float* X      = (const float*)d_in[0];
    const int*   labels = (const int*)d_in[1];
    const int N = in_sizes[1];                        // 262144

    char*  ws         = (char*)d_ws;
    int*   anchor_idx = (int*)(ws + 0);
    int*   counts     = (int*)(ws + 256);
    float* sums       = (float*)(ws + 512);
    float* anchors    = (float*)(ws + 1024);

    k_init  <<<1, 64, 0, stream>>>(anchor_idx, counts, sums);
    k_stats <<<(N + 255) / 256, 256, 0, stream>>>(labels, N, anchor_idx, counts);
    k_gather<<<NUM_CLASSES, 256, 0, stream>>>(X, anchor_idx, N, anchors);
    k_main  <<<1024, 256, SMEM_BYTES, stream>>>(X, labels, N, anchors, sums);
    k_final <<<1, 64, 0, stream>>>(sums, counts, (float*)d_out);
}